// TanhNewtonImplicitLayer_25786983645355
// MI455X (gfx1250) — compile-verified
//
#include <hip/hip_runtime.h>
#include <hip/hip_bf16.h>

// TanhNewtonImplicitLayer for MI455X (gfx1250).
//
// Reference solves z = tanh(z @ W.T + x) with Newton (batched 256x256 solves),
// then returns tanh(z* @ W.T + x) == z*. We compute the same fixed point with
// Picard iteration (the map is a contraction for W ~ U(-1/16,1/16)): each block
// owns 16 batch rows and iterates entirely in LDS -- rows are independent, so
// no inter-block sync is ever needed.
//
// CDNA5 specifics:
//  - V_WMMA_F32_16X16X4_F32: full-fp32 matrix pipe (tolerance ~2e-7/element
//    rules out f16/bf16 operands). Two independent accumulator chains per wave
//    to hide WMMA latency (4 waves/SIMD32 x 2 chains = 8 in flight).
//  - 320KB WGP LDS: W (256KB, padded rows) + ping-pong z tiles + x tile
//    = ~314KB resident; one s_barrier per iteration (ping-pong removes the
//    read/overwrite hazard barrier).
//  - global_prefetch_b8 via __builtin_prefetch for the one-time W pull.

typedef __attribute__((ext_vector_type(2))) float v2f;
typedef __attribute__((ext_vector_type(8))) float v8f;

#define NFEAT   256
#define ROWS_PB 16          // batch rows per block (= WMMA M)
#define WSTRIDE 258         // LDS row stride in floats (+2 pad breaks 256-stride bank aliasing)
#define XSTRIDE 260
#define NITER   256         // contraction ~0.4 => error << fp32 eps at the fixed point
#define THREADS 512         // 16 wave32s: one 16-col output tile per wave

__global__ void tanh_fixed_point_kernel(const float* __restrict__ x,
                                        const float* __restrict__ W,
                                        float* __restrict__ out) {
  extern __shared__ float smem[];
  float* w_lds = smem;                            // [256][WSTRIDE]  w_lds[n][k] = W[n][k]
  float* z_a   = w_lds + NFEAT * WSTRIDE;         // [16][WSTRIDE]  ping
  float* z_b   = z_a + ROWS_PB * WSTRIDE;         // [16][WSTRIDE]  pong
  float* x_lds = z_b + ROWS_PB * WSTRIDE;         // [16][XSTRIDE]

  const int tid  = threadIdx.x;
  const int row0 = blockIdx.x * ROWS_PB;

  // ---- Prefetch W toward this WGP (gfx1250: global_prefetch_b8, one per 256B line) ----
  for (int line = tid; line < (NFEAT * NFEAT) / 64; line += THREADS)
    __builtin_prefetch(W + line * 64, 0, 3);

  // ---- Load W into LDS (float2 granularity keeps 8B-aligned ds stores w/ padded stride) ----
  for (int i = tid; i < (NFEAT * NFEAT) / 2; i += THREADS) {
    const int r  = i >> 7;          // 128 float2 per row
    const int c2 = (i & 127) * 2;
    v2f v = *(const v2f*)(W + r * NFEAT + c2);
    *(v2f*)(w_lds + r * WSTRIDE + c2) = v;
  }
  // ---- Load x tile, init z0 = tanh(x) (same init as reference) ----
  for (int i = tid; i < ROWS_PB * NFEAT; i += THREADS) {
    const int r = i >> 8;
    const int c = i & 255;
    const float xv = x[(row0 + r) * NFEAT + c];
    x_lds[r * XSTRIDE + c] = xv;
    z_a[r * WSTRIDE + c] = tanhf(xv);
  }

  // ---- Per-lane WMMA fragment coordinates (wave32) ----
  const int lane = tid & 31;
  const int wave = tid >> 5;        // 0..15
  const int l15  = lane & 15;
  const int hi   = lane >> 4;       // 0: lanes 0-15, 1: lanes 16-31
  const int n0   = wave * 16;       // this wave's 16 output columns

  // A (16x4 f32): lane holds z[m=l15][k0 + 2*hi + {0,1}]                      -> float2
  // B (4x16 f32): lane holds B[k0+2*hi+{0,1}][n] = W[n0+l15][k0+2*hi+{0,1}]   -> float2
  const int aOff = l15 * WSTRIDE + 2 * hi;
  const float* bBase = w_lds + (n0 + l15) * WSTRIDE + 2 * hi;

  __syncthreads();                   // init fill visible to all waves

  for (int it = 0; it < NITER; ++it) {
    const float* zr = (it & 1) ? z_b : z_a;   // read buffer
    float*       zw = (it & 1) ? z_a : z_b;   // write buffer
    const float* aBase = zr + aOff;

    // Two independent D->C chains (even/odd K-steps) to double matrix-pipe ILP.
    v8f acc0 = {};
    v8f acc1 = {};
#pragma unroll
    for (int k0 = 0; k0 < NFEAT; k0 += 8) {
      v2f a0 = *(const v2f*)(aBase + k0);
      v2f b0 = *(const v2f*)(bBase + k0);
      acc0 = __builtin_amdgcn_wmma_f32_16x16x4_f32(
          false, a0, false, b0, (short)0, acc0, false, false);
      v2f a1 = *(const v2f*)(aBase + k0 + 4);
      v2f b1 = *(const v2f*)(bBase + k0 + 4);
      acc1 = __builtin_amdgcn_wmma_f32_16x16x4_f32(
          false, a1, false, b1, (short)0, acc1, false, false);
    }

    // C/D layout: vgpr i -> (M = i + 8*hi, N = l15). Writes go to the *other*
    // buffer, so no barrier is needed between the K-loop reads and these writes.
#pragma unroll
    for (int i = 0; i < 8; ++i) {
      const int m = i + 8 * hi;
      const int n = n0 + l15;
      zw[m * WSTRIDE + n] = tanhf(acc0[i] + acc1[i] + x_lds[m * XSTRIDE + n]);
    }
    __syncthreads();                 // zw complete before it becomes zr
  }

  // Last write (it = NITER-1) landed in z_a for even NITER.
  const float* zfin = (NITER % 2 == 0) ? z_a : z_b;

  // zfin holds tanh(z_{K-1} W^T + x) == reference's returned tensor.
  for (int i = tid; i < ROWS_PB * NFEAT; i += THREADS) {
    const int r = i >> 8;
    const int c = i & 255;
    out[(row0 + r) * NFEAT + c] = zfin[r * WSTRIDE + c];
  }
}

extern "C" void kernel_launch(void* const* d_in, const int* in_sizes, int n_in,
                              void* d_out, int out_size, void* d_ws, size_t ws_size,
                              hipStream_t stream) {
  const float* x = (const float*)d_in[0];   // [B, 256] f32
  const float* W = (const float*)d_in[1];   // [256, 256] f32 row-major
  float* out = (float*)d_out;               // [B, 256] f32

  const int B = in_sizes[0] / NFEAT;        // 1024
  const int grid = B / ROWS_PB;             // 64 blocks, one per 16 rows

  const size_t shmem =
      (size_t)(NFEAT * WSTRIDE + 2 * ROWS_PB * WSTRIDE + ROWS_PB * XSTRIDE) *
      sizeof(float);                        // ~314KB, fits 320KB WGP LDS

  // Allow >default dynamic LDS (WGP has 320KB). Not a stream op; capture-safe.
  (void)hipFuncSetAttribute((const void*)tanh_fixed_point_kernel,
                            hipFuncAttributeMaxDynamicSharedMemorySize, (int)shmem);

  tanh_fixed_point_kernel<<<grid, THREADS, shmem, stream>>>(x, W, out);
}